// LegalGAT_74491912782052
// MI455X (gfx1250) — compile-verified
//
#include <hip/hip_runtime.h>
#include <math.h>

#define N_NODES 50000
#define N_EDGES 800000
#define ETOT    850000      // edges + self loops
#define F_IN    256
#define HID     64
#define HEADS   4
#define HC1     256         // HEADS*HID
#define NEG_SLOPE 0.2f

typedef __attribute__((ext_vector_type(2))) float v2f;
typedef __attribute__((ext_vector_type(8))) float v8f;

// ---------------------------------------------------------------------------
// C[M, NCOL] = A[M, 256] * B[256, NCOL], fp32 exact via V_WMMA_F32_16X16X4_F32.
// grid.x = M/16 row tiles; block = 32 * (NCOL/16) threads; wave w owns col tile w.
// A-frag (16x4): lane<16 -> {A[m][k],A[m][k+1]}, lane>=16 -> {A[m][k+2],A[m][k+3]}
// B-frag (4x16): lane<16 -> {B[k][n],B[k+1][n]},  lane>=16 -> {B[k+2][n],B[k+3][n]}
// C/D (16x16):   VGPR r, lanes 0-15 -> row r, lanes 16-31 -> row r+8, col = lane&15
// ---------------------------------------------------------------------------
template<int NCOL>
__global__ void gemm_wmma_f32(const float* __restrict__ A,
                              const float* __restrict__ B,
                              float* __restrict__ C)
{
    const int K = 256;
    const int lane  = threadIdx.x & 31;
    const int wave  = threadIdx.x >> 5;      // column tile within block
    const int mtile = blockIdx.x;
    const int ml  = lane & 15;
    const int kh  = lane >> 4;               // 0 or 1 (K half)
    const int col = wave * 16 + ml;

    const float* arow = A + (size_t)(mtile * 16 + ml) * K;

    v8f acc = {0.f, 0.f, 0.f, 0.f, 0.f, 0.f, 0.f, 0.f};
#pragma unroll 4
    for (int k = 0; k < K; k += 4) {
        v2f a = *(const v2f*)(arow + k + 2 * kh);      // 8B aligned (k even)
        v2f b;
        b.x = B[(k + 2 * kh)     * NCOL + col];
        b.y = B[(k + 2 * kh + 1) * NCOL + col];
        acc = __builtin_amdgcn_wmma_f32_16x16x4_f32(
            /*neg_a=*/false, a, /*neg_b=*/false, b,
            /*c_mod=*/(short)0, acc, /*reuse_a=*/false, /*reuse_b=*/false);
    }
#pragma unroll
    for (int r = 0; r < 8; ++r) {
        int row = mtile * 16 + r + 8 * kh;
        C[(size_t)row * NCOL + col] = acc[r];
    }
}

// ---------------------------------------------------------------------------
// al_s[n,h] = dot(xh[n,h,:], a_src[h,:]);  al_d likewise.  One thread per (n,h).
// ---------------------------------------------------------------------------
__global__ void attn_logits(const float* __restrict__ xh,
                            const float* __restrict__ a_src,
                            const float* __restrict__ a_dst,
                            float* __restrict__ al_s,
                            float* __restrict__ al_d,
                            int heads)
{
    int t = blockIdx.x * blockDim.x + threadIdx.x;
    if (t >= N_NODES * heads) return;
    int n = t / heads;
    int h = t - n * heads;
    const float* row = xh + (size_t)n * heads * HID + h * HID;
    const float* as  = a_src + h * HID;
    const float* ad  = a_dst + h * HID;
    float ss = 0.f, sd = 0.f;
#pragma unroll 8
    for (int c = 0; c < HID; ++c) {
        float v = row[c];
        ss += v * as[c];
        sd += v * ad[c];
    }
    al_s[t] = ss;
    al_d[t] = sd;
}

// Monotone float <-> uint key so atomicMax on uint == float max (handles negatives).
__device__ __forceinline__ unsigned f2key(float f) {
    unsigned b = __float_as_uint(f);
    return (b & 0x80000000u) ? ~b : (b | 0x80000000u);
}
__device__ __forceinline__ float key2f(unsigned k) {
    return __uint_as_float((k & 0x80000000u) ? (k & 0x7FFFFFFFu) : ~k);
}

__device__ __forceinline__ void edge_sd(int e, const int* ei, int& s, int& d) {
    if (e < N_EDGES) { s = ei[e]; d = ei[N_EDGES + e]; }
    else             { s = e - N_EDGES; d = s; }          // self loop
}

// pass 1: per-dst running max of leaky_relu(al_s[src]+al_d[dst])
__global__ void edge_max(const int* __restrict__ ei,
                         const float* __restrict__ al_s,
                         const float* __restrict__ al_d,
                         unsigned* __restrict__ mkey,
                         int heads)
{
    int t = blockIdx.x * blockDim.x + threadIdx.x;
    if (t >= ETOT * heads) return;
    int e = t / heads, h = t - e * heads;
    int s, d; edge_sd(e, ei, s, d);
    float v = al_s[s * heads + h] + al_d[d * heads + h];
    v = v > 0.f ? v : NEG_SLOPE * v;
    atomicMax(&mkey[d * heads + h], f2key(v));
}

// pass 2: ex = exp(v - max); den[dst] += ex
__global__ void edge_exp(const int* __restrict__ ei,
                         const float* __restrict__ al_s,
                         const float* __restrict__ al_d,
                         const unsigned* __restrict__ mkey,
                         float* __restrict__ ex,
                         float* __restrict__ den,
                         int heads)
{
    int t = blockIdx.x * blockDim.x + threadIdx.x;
    if (t >= ETOT * heads) return;
    int e = t / heads, h = t - e * heads;
    int s, d; edge_sd(e, ei, s, d);
    float v = al_s[s * heads + h] + al_d[d * heads + h];
    v = v > 0.f ? v : NEG_SLOPE * v;
    float m  = key2f(mkey[d * heads + h]);
    float vx = expf(v - m);
    ex[t] = vx;
    atomicAdd(&den[d * heads + h], vx);
}

// pass 3: out[dst] += alpha * xh[src].  One wave32 per edge; 2 feats/lane/head.
__global__ void edge_agg(const int* __restrict__ ei,
                         const float* __restrict__ ex,
                         const float* __restrict__ den,
                         const float* __restrict__ xh,
                         float* __restrict__ out,
                         int heads)
{
    int gid  = blockIdx.x * blockDim.x + threadIdx.x;
    int lane = gid & 31;
    int e    = gid >> 5;
    if (e >= ETOT) return;
    int s, d; edge_sd(e, ei, s, d);
    for (int h = 0; h < heads; ++h) {
        float alpha = ex[e * heads + h] / (den[d * heads + h] + 1e-16f);
        const float* src = xh  + (size_t)s * heads * HID + h * HID;
        float*       dst = out + (size_t)d * heads * HID + h * HID;
        atomicAdd(&dst[lane],      alpha * src[lane]);
        atomicAdd(&dst[lane + 32], alpha * src[lane + 32]);
    }
}

// x = elu(x + bias[col]);  cols is a power of two (256 or 64), mask = cols-1
__global__ void bias_elu(float* __restrict__ x,
                         const float* __restrict__ bias,
                         int mask, int total)
{
    int t = blockIdx.x * blockDim.x + threadIdx.x;
    if (t >= total) return;
    float v = x[t] + bias[t & mask];
    x[t] = v > 0.f ? v : (expf(v) - 1.f);
}

extern "C" void kernel_launch(void* const* d_in, const int* in_sizes, int n_in,
                              void* d_out, int out_size, void* d_ws, size_t ws_size,
                              hipStream_t stream)
{
    const float* x   = (const float*)d_in[0];
    const int*   ei  = (const int*)  d_in[1];
    const float* W1  = (const float*)d_in[2];
    const float* as1 = (const float*)d_in[3];
    const float* ad1 = (const float*)d_in[4];
    const float* b1  = (const float*)d_in[5];
    const float* W2  = (const float*)d_in[6];
    const float* as2 = (const float*)d_in[7];
    const float* ad2 = (const float*)d_in[8];
    const float* b2  = (const float*)d_in[9];
    float* out = (float*)d_out;

    // workspace layout (floats)
    float* ws = (float*)d_ws;
    size_t o = 0;
    float*    xh1  = ws + o;              o += (size_t)N_NODES * HC1;   // x @ W1
    float*    h1   = ws + o;              o += (size_t)N_NODES * HC1;   // agg accumulator -> elu'd hidden
    float*    xh2  = ws + o;              o += (size_t)N_NODES * HID;   // h1 @ W2
    float*    als1 = ws + o;              o += (size_t)N_NODES * HEADS;
    float*    ald1 = ws + o;              o += (size_t)N_NODES * HEADS;
    unsigned* km1  = (unsigned*)(ws + o); o += (size_t)N_NODES * HEADS;
    float*    den1 = ws + o;              o += (size_t)N_NODES * HEADS;
    float*    ex1  = ws + o;              o += (size_t)ETOT    * HEADS;
    float*    als2 = ws + o;              o += (size_t)N_NODES;
    float*    ald2 = ws + o;              o += (size_t)N_NODES;
    unsigned* km2  = (unsigned*)(ws + o); o += (size_t)N_NODES;
    float*    den2 = ws + o;              o += (size_t)N_NODES;
    float*    ex2  = ws + o;              o += (size_t)ETOT;

    (void)hipMemsetAsync(h1,   0, (size_t)N_NODES * HC1   * sizeof(float), stream);
    (void)hipMemsetAsync(km1,  0, (size_t)N_NODES * HEADS * sizeof(unsigned), stream);
    (void)hipMemsetAsync(den1, 0, (size_t)N_NODES * HEADS * sizeof(float), stream);
    (void)hipMemsetAsync(km2,  0, (size_t)N_NODES         * sizeof(unsigned), stream);
    (void)hipMemsetAsync(den2, 0, (size_t)N_NODES         * sizeof(float), stream);
    (void)hipMemsetAsync(out,  0, (size_t)N_NODES * HID   * sizeof(float), stream);

    const int EB = 256;

    // ---- layer 1 (heads=4, concat) ----
    gemm_wmma_f32<HC1><<<dim3(N_NODES / 16), dim3(32 * (HC1 / 16)), 0, stream>>>(x, W1, xh1);
    attn_logits<<<(N_NODES * HEADS + EB - 1) / EB, EB, 0, stream>>>(xh1, as1, ad1, als1, ald1, HEADS);
    edge_max   <<<(ETOT * HEADS + EB - 1) / EB, EB, 0, stream>>>(ei, als1, ald1, km1, HEADS);
    edge_exp   <<<(ETOT * HEADS + EB - 1) / EB, EB, 0, stream>>>(ei, als1, ald1, km1, ex1, den1, HEADS);
    edge_agg   <<<((size_t)ETOT * 32 + EB - 1) / EB, EB, 0, stream>>>(ei, ex1, den1, xh1, h1, HEADS);
    bias_elu   <<<((size_t)N_NODES * HC1 + EB - 1) / EB, EB, 0, stream>>>(h1, b1, HC1 - 1, N_NODES * HC1);

    // ---- layer 2 (heads=1, mean == identity) ----
    gemm_wmma_f32<HID><<<dim3(N_NODES / 16), dim3(32 * (HID / 16)), 0, stream>>>(h1, W2, xh2);
    attn_logits<<<(N_NODES + EB - 1) / EB, EB, 0, stream>>>(xh2, as2, ad2, als2, ald2, 1);
    edge_max   <<<(ETOT + EB - 1) / EB, EB, 0, stream>>>(ei, als2, ald2, km2, 1);
    edge_exp   <<<(ETOT + EB - 1) / EB, EB, 0, stream>>>(ei, als2, ald2, km2, ex2, den2, 1);
    edge_agg   <<<((size_t)ETOT * 32 + EB - 1) / EB, EB, 0, stream>>>(ei, ex2, den2, xh2, out, 1);
    bias_elu   <<<((size_t)N_NODES * HID + EB - 1) / EB, EB, 0, stream>>>(out, b2, HID - 1, N_NODES * HID);
}